// VectorQuantizer_82729660056146
// MI455X (gfx1250) — compile-verified
//
#include <hip/hip_runtime.h>
#include <hip/hip_bf16.h>

// ---------------------------------------------------------------------------
// VQ-VAE vector quantizer forward for MI455X (gfx1250, wave32).
//   z:        [16,2048,128] fp32  (M=32768 rows, D=128)
//   codebook: [1024,128]    fp32  (K=1024 entries)
// d_out (fp32): z_q [M*128] | indices [M] | commit | cb_loss | perplexity | dist [M*1024]
//
// Pipeline: TDM (tensor_load_to_lds) double-buffers padded codebook chunks in
// LDS while all 8 waves run V_WMMA_F32_16X16X4_F32 distance GEMM tiles.
// ---------------------------------------------------------------------------

typedef __attribute__((ext_vector_type(2))) float v2f;
typedef __attribute__((ext_vector_type(8))) float v8f;
typedef unsigned int u32x4 __attribute__((ext_vector_type(4)));
typedef int i32x4 __attribute__((ext_vector_type(4)));
typedef int i32x8 __attribute__((ext_vector_type(8)));

#define DIMD 128
#define KCB  1024
#define MTOT 32768
#define N_CH 128            // codebook entries per LDS chunk
#define NPAD 132            // padded row stride in dwords ([n][k] layout, conflict-free)
#define NCHUNK (KCB / N_CH) // 8
#define WAVES 8
#define ROWS_PER_BLOCK (WAVES * 16)   // 128 rows/block -> 256 blocks

#define ZQ_OFF     0
#define IDX_OFF    4194304              // 32768*128
#define COMMIT_OFF 4227072              // + 32768
#define CBL_OFF    4227073
#define PERP_OFF   4227074
#define DIST_OFF   4227075

// ---- TDM: stage one codebook chunk (128 entries x 128 f32) into LDS,
// row-major with 4-dword padding every 128 dwords -> NPAD=132 stride.
__device__ __forceinline__ void tdm_load_chunk(const float* gsrc, unsigned ldsByte) {
  unsigned long long ga = (unsigned long long)(size_t)gsrc;
  u32x4 g0 = { 1u,                                  // count=1 (valid user D#)
               ldsByte,                              // lds_addr
               (unsigned)(ga & 0xFFFFFFFFull),       // global_addr[31:0]
               (unsigned)((ga >> 32) & 0x01FFFFFFull) | (2u << 30) }; // ga[56:32] | type=2
  i32x8 g1 = { (2 << 16) | (1 << 20) | (6 << 22) | (3 << 25),
               //  data_size=4B | pad_enable | pad_interval=128dw | pad_amount=4dw
               (int)(128u << 16),   // tensor_dim0 = 128 (elements per row)
               (int)(1024u << 16),  // tensor_dim1 = 1024 rows
               (int)(128u << 16),   // tile_dim0 = 128
               128,                 // tile_dim1 = 128 rows per chunk
               128,                 // tensor_dim0_stride = 128
               0, 0 };
  i32x4 g2 = { 1, 1, 0, 0 };               // tensor_dim2=1, tensor_dim3=1 (unused dims)
  i32x4 g3 = { 0, (int)(1u << 16), 0, 0 }; // tensor_dim4=1
  i32x8 g4 = { 0, 0, 0, 0, 0, 0, 0, 0 };   // trailing group (clang-23 6-arg form)
  __builtin_amdgcn_tensor_load_to_lds(g0, g1, g2, g3, g4, 0);
}

// ---------------- init: codebook norms + zero counts ------------------------
__global__ void vq_init(const float* __restrict__ cb, float* __restrict__ cnorm,
                        unsigned int* __restrict__ counts) {
  int k = blockIdx.x * blockDim.x + threadIdx.x;
  if (k < KCB) {
    const float* row = cb + (size_t)k * DIMD;
    float s = 0.f;
    for (int j = 0; j < DIMD; j += 4) {
      float4 v = *(const float4*)(row + j);
      s += v.x * v.x + v.y * v.y + v.z * v.z + v.w * v.w;
    }
    cnorm[k] = s;
    counts[k] = 0u;
  }
}

// ---------------- main fused kernel -----------------------------------------
__global__ __launch_bounds__(256) void vq_main(
    const float* __restrict__ z, const float* __restrict__ cb,
    const float* __restrict__ cnorm, float* __restrict__ out,
    float* __restrict__ blockSums, unsigned int* __restrict__ counts) {

  __shared__ __attribute__((aligned(128))) float ldsB[2][N_CH * NPAD]; // 2 x 67.6 KB
  __shared__ float ldsRed[WAVES];

  const int tid  = threadIdx.x;
  const int wave = tid >> 5;
  const int lane = tid & 31;
  const int half = lane >> 4;
  const int lid  = lane & 15;

  const int m0 = blockIdx.x * ROWS_PER_BLOCK + wave * 16;

  // ---- A operand: 16x128 in V_WMMA_F32_16X16X4_F32 layout.
  // a[kk].x : K = 4kk + 2*half ; a[kk].y : K = 4kk + 2*half + 1 ; M = lid.
  v2f a[32];
  float pn = 0.f;
  {
    const float* zrow = z + (size_t)(m0 + lid) * DIMD;
#pragma unroll
    for (int kk = 0; kk < 32; ++kk) {
      float x0 = zrow[kk * 4 + 2 * half + 0];
      float x1 = zrow[kk * 4 + 2 * half + 1];
      a[kk].x = x0; a[kk].y = x1;
      pn += x0 * x0 + x1 * x1;
    }
  }
  pn += __shfl_xor(pn, 16, 32);        // every lane: ||z_row(lid)||^2
  float rn[8];
#pragma unroll
  for (int i = 0; i < 8; ++i) rn[i] = __shfl(pn, half * 8 + i, 32);

  float bestD[8];
  int   bestN[8];
#pragma unroll
  for (int i = 0; i < 8; ++i) { bestD[i] = 3.4e38f; bestN[i] = 0; }

  float* distOut = out + DIST_OFF;

  // ---- prologue: TDM chunk 0 into buffer 0 (one DMA per block, wave 0 only)
  if (wave == 0) tdm_load_chunk(cb, (unsigned)(size_t)&ldsB[0][0]);

  for (int c = 0; c < NCHUNK; ++c) {
    __builtin_amdgcn_s_wait_tensorcnt(0);   // wave0: chunk c landed; others: no-op
    __syncthreads();
    if (wave == 0 && c + 1 < NCHUNK)        // overlap: DMA chunk c+1 while computing c
      tdm_load_chunk(cb + (size_t)(c + 1) * N_CH * DIMD,
                     (unsigned)(size_t)&ldsB[(c + 1) & 1][0]);

    const float* bufB = &ldsB[c & 1][0];

#pragma unroll
    for (int nt = 0; nt < N_CH / 16; ++nt) {
      v8f acc = {};
      const int nb = nt * 16 + lid;
      // per-lane base: entry nb, dword 2*half -> all 32 slabs at immediate offsets
      const float* bbase = bufB + nb * NPAD + 2 * half;
#pragma unroll
      for (int g = 0; g < 4; ++g) {
        v2f bv[8];                        // burst 8 ds_load_b64, then 8 WMMAs
#pragma unroll
        for (int j = 0; j < 8; ++j) bv[j] = *(const v2f*)(bbase + (g * 8 + j) * 4);
#pragma unroll
        for (int j = 0; j < 8; ++j)
          acc = __builtin_amdgcn_wmma_f32_16x16x4_f32(
              false, a[g * 8 + j], false, bv[j], (short)0, acc, false, false);
      }

      const int n  = c * N_CH + nb;
      const float cn = cnorm[n];
#pragma unroll
      for (int i = 0; i < 8; ++i) {       // C/D row M = i + 8*half, col = lid
        const int mrow = m0 + half * 8 + i;
        float dist = rn[i] - 2.f * acc[i] + cn;
        distOut[(size_t)mrow * KCB + n] = dist;
        if (dist < bestD[i]) { bestD[i] = dist; bestN[i] = n; }
      }
    }
  }

  // ---- wave-local argmin across the 16 lanes of each half (xor butterfly)
#pragma unroll
  for (int off = 1; off < 16; off <<= 1) {
#pragma unroll
    for (int i = 0; i < 8; ++i) {
      float dO = __shfl_xor(bestD[i], off, 32);
      int   nO = __shfl_xor(bestN[i], off, 32);
      if (dO < bestD[i] || (dO == bestD[i] && nO < bestN[i])) {
        bestD[i] = dO; bestN[i] = nO;
      }
    }
  }

  // ---- indices (fp32, harness output dtype) + code-usage counts
  if (lid == 0) {
#pragma unroll
    for (int i = 0; i < 8; ++i) {
      int mrow = m0 + half * 8 + i;
      out[IDX_OFF + mrow] = (float)bestN[i];
      atomicAdd(&counts[bestN[i]], 1u);
    }
  }

  // ---- broadcast all 16 row winners to the whole wave
  int idxRow[16];
#pragma unroll
  for (int r = 0; r < 16; ++r)
    idxRow[r] = __shfl(bestN[r & 7], (r >> 3) * 16, 32);

  // ---- z_q gather + MSE partial (deterministic per-block slot)
  float ls = 0.f;
  const int k4 = lane * 4;
#pragma unroll 4
  for (int r = 0; r < 16; ++r) {
    const int mrow = m0 + r;
    const float4 q  = *(const float4*)(cb + (size_t)idxRow[r] * DIMD + k4);
    *(float4*)(out + ZQ_OFF + (size_t)mrow * DIMD + k4) = q;
    const float4 zv = *(const float4*)(z + (size_t)mrow * DIMD + k4);
    float dx = q.x - zv.x, dy = q.y - zv.y, dz = q.z - zv.z, dw = q.w - zv.w;
    ls += dx * dx + dy * dy + dz * dz + dw * dw;
  }
#pragma unroll
  for (int off = 16; off >= 1; off >>= 1) ls += __shfl_xor(ls, off, 32);
  if (lane == 0) ldsRed[wave] = ls;
  __syncthreads();
  if (tid == 0) {
    float s = 0.f;
    for (int w = 0; w < WAVES; ++w) s += ldsRed[w];
    blockSums[blockIdx.x] = s;
  }
}

// ---------------- finalize: losses + perplexity -----------------------------
__global__ __launch_bounds__(1024) void vq_finalize(
    const unsigned int* __restrict__ counts, const float* __restrict__ blockSums,
    float* __restrict__ out) {
  __shared__ float redP[32];
  __shared__ float redS[32];
  const int t = threadIdx.x, wv = t >> 5, ln = t & 31;

  float p = (float)counts[t] * (1.0f / (float)MTOT);
  float term = p * __logf(p + 1e-10f);
  float bs = (t < 256) ? blockSums[t] : 0.f;
#pragma unroll
  for (int off = 16; off >= 1; off >>= 1) {
    term += __shfl_xor(term, off, 32);
    bs   += __shfl_xor(bs, off, 32);
  }
  if (ln == 0) { redP[wv] = term; redS[wv] = bs; }
  __syncthreads();
  if (t < 32) {
    float sp = redP[t], ss = redS[t];
#pragma unroll
    for (int off = 16; off >= 1; off >>= 1) {
      sp += __shfl_xor(sp, off, 32);
      ss += __shfl_xor(ss, off, 32);
    }
    if (t == 0) {
      float cbl = ss * (1.0f / (float)((size_t)MTOT * DIMD));
      out[CBL_OFF]    = cbl;
      out[COMMIT_OFF] = 0.25f * cbl;
      out[PERP_OFF]   = __expf(-sp);
    }
  }
}

// ---------------------------------------------------------------------------
extern "C" void kernel_launch(void* const* d_in, const int* in_sizes, int n_in,
                              void* d_out, int out_size, void* d_ws, size_t ws_size,
                              hipStream_t stream) {
  const float* z  = (const float*)d_in[0];
  const float* cb = (const float*)d_in[1];
  float* out = (float*)d_out;

  float*        cnorm     = (float*)d_ws;                          // 1024 f32
  unsigned int* counts    = (unsigned int*)((char*)d_ws + 4096);   // 1024 u32
  float*        blockSums = (float*)((char*)d_ws + 8192);          // 256 f32

  vq_init<<<4, 256, 0, stream>>>(cb, cnorm, counts);
  vq_main<<<MTOT / ROWS_PER_BLOCK, 256, 0, stream>>>(z, cb, cnorm, out, blockSums, counts);
  vq_finalize<<<1, 1024, 0, stream>>>(counts, blockSums, out);
}